// TransitionUp_39728447488679
// MI455X (gfx1250) — compile-verified
//
#include <hip/hip_runtime.h>
#include <hip/hip_bf16.h>
#include <cstdint>

typedef __attribute__((ext_vector_type(2))) float v2f;
typedef __attribute__((ext_vector_type(8))) float v8f;

__device__ __forceinline__ v8f wmma_f32x4(v2f a, v2f b, v8f c) {
  // D = A(16x4) * B(4x16) + C(16x16), fp32
  return __builtin_amdgcn_wmma_f32_16x16x4_f32(
      /*neg_a=*/false, a, /*neg_b=*/false, b,
      /*c_mod=*/(short)0, c, /*reuse_a=*/false, /*reuse_b=*/false);
}

#define NEG_SLOPE 0.2f
#define VN_EPS    1e-6f
#define DIST_EPS  1e-8f

// ---------------------------------------------------------------------------
// VN Linear + LeakyReLU fused via WMMA f32 16x16x4.
// X: [CIN][3][Npts] row-major.  Wf/Wd: [64][CIN] row-major.
// One wave: 16 output rows x 16 points x all 3 v-planes -> 6 accumulators;
// VN reflection done per-lane in registers (no cross-lane traffic needed
// thanks to the C/D fragment layout: row = r + 8*(lane>=16), col = lane%16).
// NPTS != 0 specializes the point count so all B-fragment addressing
// strength-reduces to immediate offsets. Distance-1 software pipeline
// overlaps next fragment loads with current wmmas.
// ---------------------------------------------------------------------------
template<int CIN, bool WS_LAYOUT, int NPTS>
__global__ __launch_bounds__(256) void vn_linear_lrelu_wmma(
    const float* __restrict__ X,
    const float* __restrict__ Wf,
    const float* __restrict__ Wd,
    float* __restrict__ out,
    int NptsArg)
{
  const int Np = NPTS ? NPTS : NptsArg;

  const int lane  = threadIdx.x & 31;
  const int wv    = threadIdx.x >> 5;        // 8 waves / block
  const int lm    = lane & 15;               // point within tile
  const int hh    = lane >> 4;               // half-wave (K / row split)
  const int otile = wv & 3;                  // 4 o-tiles cover Cout=64
  const int ptile = blockIdx.x * 2 + (wv >> 2);
  const int ob    = otile * 16;
  const int nb    = ptile * 16;

  const size_t colStride = (size_t)3 * (size_t)Np;  // X row stride (floats)

  v8f accP[3], accD[3];
  const v8f zero = {0.f,0.f,0.f,0.f,0.f,0.f,0.f,0.f};
#pragma unroll
  for (int v = 0; v < 3; ++v) { accP[v] = zero; accD[v] = zero; }

  const float* wfp   = Wf + (size_t)(ob + lm) * CIN + 2 * hh;
  const float* wdp   = Wd + (size_t)(ob + lm) * CIN + 2 * hh;
  const float* xbase = X + (size_t)nb + lm + (size_t)(2 * hh) * colStride;

  auto loadA = [&](int kb, v2f& aF, v2f& aD) {
    aF.x = wfp[kb]; aF.y = wfp[kb + 1];      // contiguous -> b64 load
    aD.x = wdp[kb]; aD.y = wdp[kb + 1];
  };
  auto loadB = [&](int kb, v2f& b0, v2f& b1, v2f& b2) {
    const float* xk = xbase + (size_t)kb * colStride;
    b0.x = xk[0];                  b0.y = xk[colStride];
    b1.x = xk[(size_t)Np];         b1.y = xk[(size_t)Np + colStride];
    b2.x = xk[(size_t)2 * Np];     b2.y = xk[(size_t)2 * Np + colStride];
  };

  v2f aF, aD, b0, b1, b2;
  loadA(0, aF, aD);
  loadB(0, b0, b1, b2);

#pragma unroll
  for (int kb = 0; kb < CIN; kb += 4) {
    v2f naF = aF, naD = aD, nb0 = b0, nb1 = b1, nb2 = b2;
    if (kb + 4 < CIN) {                       // compile-time under unroll
      loadA(kb + 4, naF, naD);
      loadB(kb + 4, nb0, nb1, nb2);
    }
    if (kb + 16 < CIN) {                      // warm L2 ahead of the stream
      __builtin_prefetch(xbase + (size_t)(kb + 16) * colStride, 0, 1);
    }
    accP[0] = wmma_f32x4(aF, b0, accP[0]);
    accD[0] = wmma_f32x4(aD, b0, accD[0]);
    accP[1] = wmma_f32x4(aF, b1, accP[1]);
    accD[1] = wmma_f32x4(aD, b1, accD[1]);
    accP[2] = wmma_f32x4(aF, b2, accP[2]);
    accD[2] = wmma_f32x4(aD, b2, accD[2]);
    aF = naF; aD = naD; b0 = nb0; b1 = nb1; b2 = nb2;
  }

  // VN leaky-ReLU combine: lane holds p/d for point pt, rows ob+r+8*hh, v=0..2
  const int pt = nb + lm;
#pragma unroll
  for (int r = 0; r < 8; ++r) {
    const int o = ob + r + 8 * hh;
    float p0 = accP[0][r], p1 = accP[1][r], p2 = accP[2][r];
    float d0 = accD[0][r], d1 = accD[1][r], d2 = accD[2][r];
    float dot = p0 * d0 + p1 * d1 + p2 * d2;
    float dd  = d0 * d0 + d1 * d1 + d2 * d2;
    float s   = dot / (dd + VN_EPS);
    bool neg  = dot < 0.0f;
    float q0 = neg ? (p0 - s * d0) : p0;
    float q1 = neg ? (p1 - s * d1) : p1;
    float q2 = neg ? (p2 - s * d2) : p2;
    float y0 = NEG_SLOPE * p0 + (1.0f - NEG_SLOPE) * q0;
    float y1 = NEG_SLOPE * p1 + (1.0f - NEG_SLOPE) * q1;
    float y2 = NEG_SLOPE * p2 + (1.0f - NEG_SLOPE) * q2;
    if constexpr (WS_LAYOUT) {
      float* op = out + (size_t)pt * 192 + (size_t)o * 3;
      op[0] = y0; op[1] = y1; op[2] = y2;
    } else {
      out[((size_t)(o * 3 + 0)) * (size_t)Np + pt] = y0;
      out[((size_t)(o * 3 + 1)) * (size_t)Np + pt] = y1;
      out[((size_t)(o * 3 + 2)) * (size_t)Np + pt] = y2;
    }
  }
}

// ---------------------------------------------------------------------------
// kNN (k=3), brute force with LDS-staged candidate cloud packed as float4
// (one ds_load_b128 per candidate; 64 KB LDS, well under 320 KB/WGP).
// One thread per query; sorted top-3 kept in registers.
// ---------------------------------------------------------------------------
#define MAXN2 4096

__global__ __launch_bounds__(256) void knn3_kernel(
    const float* __restrict__ p1, const float* __restrict__ p2,
    int n1, int n2, int blocksPerBatch,
    int* __restrict__ oidx, float* __restrict__ ow)
{
  __shared__ float4 sp[MAXN2];
  const int batch = blockIdx.x / blocksPerBatch;
  const int qbase = (blockIdx.x % blocksPerBatch) * blockDim.x;

  for (int j = threadIdx.x; j < n2; j += blockDim.x) {
    const float* pp = p2 + (size_t)(batch * n2 + j) * 3;
    sp[j] = make_float4(pp[0], pp[1], pp[2], 0.0f);
  }
  __syncthreads();

  const int q  = qbase + threadIdx.x;
  const int gq = batch * n1 + q;
  const float qx = p1[(size_t)gq * 3 + 0];
  const float qy = p1[(size_t)gq * 3 + 1];
  const float qz = p1[(size_t)gq * 3 + 2];

  float b0 = 3.4e38f, b1 = 3.4e38f, b2 = 3.4e38f;
  int   i0 = 0,       i1 = 0,       i2 = 0;

#pragma unroll 8
  for (int j = 0; j < n2; ++j) {
    float4 s = sp[j];
    float dx = qx - s.x, dy = qy - s.y, dz = qz - s.z;
    float d  = dx * dx + dy * dy + dz * dz;
    if (d < b2) {
      if (d < b1) {
        b2 = b1; i2 = i1;
        if (d < b0) { b1 = b0; i1 = i0; b0 = d; i0 = j; }
        else        { b1 = d;  i1 = j; }
      } else        { b2 = d;  i2 = j; }
    }
  }

  float w0 = 1.0f / (b0 + DIST_EPS);
  float w1 = 1.0f / (b1 + DIST_EPS);
  float w2 = 1.0f / (b2 + DIST_EPS);
  float inv = 1.0f / (w0 + w1 + w2);
  oidx[(size_t)gq * 3 + 0] = batch * n2 + i0;
  oidx[(size_t)gq * 3 + 1] = batch * n2 + i1;
  oidx[(size_t)gq * 3 + 2] = batch * n2 + i2;
  ow[(size_t)gq * 3 + 0] = w0 * inv;
  ow[(size_t)gq * 3 + 1] = w1 * inv;
  ow[(size_t)gq * 3 + 2] = w2 * inv;
}

// ---------------------------------------------------------------------------
// Inverse-distance interpolate from point-major features + add into output.
// Point index varies fastest -> coalesced output RMW; feature gathers are
// L2-resident (12.6 MB table vs 192 MB L2).
// ---------------------------------------------------------------------------
__global__ __launch_bounds__(256) void interp_add_kernel(
    const float* __restrict__ feat,   // [N2][C3]
    const int*   __restrict__ idx,    // [N1][3]
    const float* __restrict__ w,      // [N1][3]
    float* __restrict__ out,          // [C3][N1]
    int totalN1, int C3)
{
  size_t t  = (size_t)blockIdx.x * blockDim.x + threadIdx.x;
  int pt = (int)(t % (size_t)totalN1);
  int c3 = (int)(t / (size_t)totalN1);
  if (c3 >= C3) return;
  int   j0 = idx[(size_t)pt * 3 + 0];
  int   j1 = idx[(size_t)pt * 3 + 1];
  int   j2 = idx[(size_t)pt * 3 + 2];
  float w0 = w[(size_t)pt * 3 + 0];
  float w1 = w[(size_t)pt * 3 + 1];
  float w2 = w[(size_t)pt * 3 + 2];
  float val = w0 * feat[(size_t)j0 * C3 + c3]
            + w1 * feat[(size_t)j1 * C3 + c3]
            + w2 * feat[(size_t)j2 * C3 + c3];
  out[(size_t)c3 * totalN1 + pt] += val;
}

// ---------------------------------------------------------------------------
extern "C" void kernel_launch(void* const* d_in, const int* in_sizes, int n_in,
                              void* d_out, int out_size, void* d_ws, size_t ws_size,
                              hipStream_t stream) {
  const float* p1  = (const float*)d_in[0];
  const float* x1  = (const float*)d_in[1];
  const float* p2  = (const float*)d_in[3];
  const float* x2  = (const float*)d_in[4];
  const float* w1f = (const float*)d_in[6];
  const float* w1d = (const float*)d_in[7];
  const float* w2f = (const float*)d_in[8];
  const float* w2d = (const float*)d_in[9];

  const int b  = in_sizes[2];          // 4
  const int N1 = in_sizes[0] / 3;      // 65536 fine points total
  const int N2 = in_sizes[3] / 3;      // 16384 coarse points total
  const int n1 = N1 / b;
  const int n2 = N2 / b;
  const int COUT = 64;
  const int C3   = 3 * COUT;           // 192

  // Workspace layout
  float* y2ws = (float*)d_ws;                               // N2*C3 f32
  int*   kidx = (int*)(y2ws + (size_t)N2 * C3);             // N1*3 i32
  float* kw   = (float*)(kidx + (size_t)N1 * 3);            // N1*3 f32

  // 1) y2 = VNLinLReLU(x2, w2) -> point-major ws          (Cin=128)
  if (N2 == 16384)
    vn_linear_lrelu_wmma<128, true, 16384>
        <<<dim3(N2 / 32), 256, 0, stream>>>(x2, w2f, w2d, y2ws, N2);
  else
    vn_linear_lrelu_wmma<128, true, 0>
        <<<dim3(N2 / 32), 256, 0, stream>>>(x2, w2f, w2d, y2ws, N2);

  // 2) y1 = VNLinLReLU(x1, w1) -> d_out (final layout)    (Cin=64)
  if (N1 == 65536)
    vn_linear_lrelu_wmma<64, false, 65536>
        <<<dim3(N1 / 32), 256, 0, stream>>>(x1, w1f, w1d, (float*)d_out, N1);
  else
    vn_linear_lrelu_wmma<64, false, 0>
        <<<dim3(N1 / 32), 256, 0, stream>>>(x1, w1f, w1d, (float*)d_out, N1);

  // 3) per-batch 3-NN of fine points among coarse points
  const int bpb = n1 / 256;
  knn3_kernel<<<dim3(b * bpb), 256, 0, stream>>>(p1, p2, n1, n2, bpb, kidx, kw);

  // 4) out += inverse-distance interpolation of y2
  size_t tot = (size_t)C3 * (size_t)N1;
  interp_add_kernel<<<dim3((unsigned)((tot + 255) / 256)), 256, 0, stream>>>(
      y2ws, kidx, kw, (float*)d_out, N1, C3);
}